// Model_85968065397043
// MI455X (gfx1250) — compile-verified
//
#include <hip/hip_runtime.h>
#include <hip/hip_bf16.h>
#include <math.h>

// ---------------------------------------------------------------------------
// CDNA5 fp32 WMMA helpers (V_WMMA_F32_16X16X4_F32, wave32)
// ---------------------------------------------------------------------------
typedef float v2f __attribute__((ext_vector_type(2)));
typedef float v8f __attribute__((ext_vector_type(8)));

// global-address-space pointers: force global_load/global_store (not flat_*,
// which would tick DScnt and couple waits to the LDS pipe)
#define GAS __attribute__((address_space(1)))
typedef GAS const float* gcf;
typedef GAS float*       gf;
typedef GAS const v2f*   gcv2f;

__device__ __forceinline__ v8f wmma4(v2f a, v2f b, v8f c) {
    // 8 args: (neg_a, A, neg_b, B, c_mod, C, reuse_a, reuse_b)
    return __builtin_amdgcn_wmma_f32_16x16x4_f32(
        false, a, false, b, (short)0, c, false, false);
}

#define CDIV(a, b) (((a) + (b) - 1) / (b))

// ---------------------------------------------------------------------------
// Generic strided batched GEMM: C[bz] = A[bz] (MxK) * B[bz] (KxN)  (+bias)
// Requires K % 4 == 0. Loads are unguarded (indices clamped); stores guarded.
// One wave computes a 16x64 tile, straight-line K loop.
// ---------------------------------------------------------------------------
__global__ __launch_bounds__(32)
void gemm_f32_kernel(const float* __restrict__ A_, long lda_r, long lda_c, long strideA,
                     const float* __restrict__ B_, long ldb_r, long ldb_c, long strideB,
                     float* __restrict__ C_, long ldc, long strideC,
                     int M, int N, int K,
                     const float* __restrict__ bias_, int bias_mode)
{
    const int lane = threadIdx.x & 31;
    const int half = lane >> 4;      // 0: K pair (0,1); 1: K pair (2,3)
    const int l16  = lane & 15;
    const int n0 = blockIdx.x * 64;
    const int m0 = blockIdx.y * 16;

    gcf A    = (gcf)A_ + (size_t)blockIdx.z * strideA;
    gcf B    = (gcf)B_ + (size_t)blockIdx.z * strideB;
    gf  C    = (gf)C_  + (size_t)blockIdx.z * strideC;
    gcf bias = (gcf)bias_;

    const int arow = min(m0 + l16, M - 1);                 // clamped, store-guarded later
    gcf ap = A + (size_t)arow * lda_r + (size_t)(half * 2) * lda_c;

    gcf bp[4];
#pragma unroll
    for (int t = 0; t < 4; ++t) {
        const int col = min(n0 + t * 16 + l16, N - 1);     // clamped
        bp[t] = B + (size_t)(half * 2) * ldb_r + (size_t)col * ldb_c;
    }

    v8f acc[4] = {};
    for (int k0 = 0; k0 < K; k0 += 4) {
        v2f af;
        af.x = ap[0];
        af.y = ap[lda_c];
#pragma unroll
        for (int t = 0; t < 4; ++t) {
            v2f bf;
            bf.x = bp[t][0];
            bf.y = bp[t][ldb_r];
            acc[t] = wmma4(af, bf, acc[t]);
            bp[t] += 4 * ldb_r;
        }
        ap += 4 * lda_c;
    }

#pragma unroll
    for (int t = 0; t < 4; ++t) {
        const int col = n0 + t * 16 + l16;
        if (col >= N) continue;
#pragma unroll
        for (int r = 0; r < 8; ++r) {
            const int row = m0 + r + half * 8;
            if (row >= M) continue;
            float v = acc[t][r];
            if (bias_mode == 1)      v += bias[row];
            else if (bias_mode == 2) v += bias[col];
            C[(size_t)row * ldc + col] = v;
        }
    }
}

// ---------------------------------------------------------------------------
// GEMM + row-dot:  for each z: T = A (MxK, row-major) * Bz (KxN, row-major N)
//   partial[z][row][ntile] = sum_{col in wave's 64 cols} T[row,col]*R[row,col]
// Requires K % 4 == 0 and N % 64 == 0 (true: 768/256). 32x64 tile per wave,
// A fragments reused 4x, B fragments reused 2x -> 8 WMMA per 10 loads.
// Deterministic split reduction (summed by reduce_p_kernel).
// ---------------------------------------------------------------------------
__global__ __launch_bounds__(32)
void gemm_rowdot_kernel(const float* __restrict__ A_, int lda,
                        const float* __restrict__ B_, long strideB,
                        const float* __restrict__ R_, int ldr,
                        float* __restrict__ P_,
                        int M, int N, int K, int ntiles)
{
    const int lane = threadIdx.x & 31;
    const int half = lane >> 4;
    const int l16  = lane & 15;
    const int n0 = blockIdx.x * 64;
    const int m0 = blockIdx.y * 32;

    gcf A = (gcf)A_;
    gcf R = (gcf)R_;
    gf  P = (gf)P_;

    gcf ap0 = A + (size_t)min(m0 + l16,      M - 1) * lda + half * 2;
    gcf ap1 = A + (size_t)min(m0 + 16 + l16, M - 1) * lda + half * 2;
    gcf bp  = (gcf)B_ + (size_t)blockIdx.z * strideB
                      + (size_t)(half * 2) * N + n0 + l16;

    v8f acc[8] = {};
    for (int k0 = 0; k0 < K; k0 += 4) {
        v2f a0 = *(gcv2f)ap0;      // contiguous K pair -> global_load_b64
        v2f a1 = *(gcv2f)ap1;
        v2f b0, b1, b2, b3;
        b0.x = bp[0];  b0.y = bp[N];
        b1.x = bp[16]; b1.y = bp[N + 16];
        b2.x = bp[32]; b2.y = bp[N + 32];
        b3.x = bp[48]; b3.y = bp[N + 48];
        acc[0] = wmma4(a0, b0, acc[0]);
        acc[1] = wmma4(a0, b1, acc[1]);
        acc[2] = wmma4(a0, b2, acc[2]);
        acc[3] = wmma4(a0, b3, acc[3]);
        acc[4] = wmma4(a1, b0, acc[4]);
        acc[5] = wmma4(a1, b1, acc[5]);
        acc[6] = wmma4(a1, b2, acc[6]);
        acc[7] = wmma4(a1, b3, acc[7]);
        ap0 += 4; ap1 += 4;
        bp  += (size_t)4 * N;
    }

    // row-dot epilogue: reduce acc*R over this wave's 64 columns.
#pragma unroll
    for (int g = 0; g < 2; ++g) {
        float dots[8];
#pragma unroll
        for (int r = 0; r < 8; ++r) {
            const int row = m0 + g * 16 + r + half * 8;
            float v = 0.f;
            if (row < M) {
#pragma unroll
                for (int t = 0; t < 4; ++t) {
                    const int col = n0 + t * 16 + l16;
                    v += acc[g * 4 + t][r] * R[(size_t)row * ldr + col];
                }
            }
            for (int mm = 1; mm < 16; mm <<= 1) v += __shfl_xor(v, mm, 32);
            dots[r] = v;
        }
        if (l16 == 0) {
#pragma unroll
            for (int r = 0; r < 8; ++r) {
                const int row = m0 + g * 16 + r + half * 8;
                if (row < M)
                    P[((size_t)blockIdx.z * M + row) * ntiles + blockIdx.x] = dots[r];
            }
        }
    }
}

__global__ void reduce_p_kernel(const float* __restrict__ P, float* __restrict__ out,
                                int Z, int M, int nt, const float* __restrict__ bias)
{
    long idx = (long)blockIdx.x * blockDim.x + threadIdx.x;
    long total = (long)Z * M;
    if (idx >= total) return;
    int row = (int)(idx / Z);
    int z   = (int)(idx - (long)row * Z);
    float s = 0.f;
    const float* p = P + ((size_t)z * M + row) * nt;
    for (int t = 0; t < nt; ++t) s += p[t];
    if (bias) s += bias[z];
    out[idx] = s;   // out layout: [row][z]
}

// ---------------------------------------------------------------------------
// Implicit-GEMM 3x3 SAME conv on pre-padded input, tap-major weights.
//   Wt: [Cout][9][Cp] (tap-major, channel-inner), Cp % 4 == 0
//   Xp: [B][Cp][H+2][W+2], zero border & zero extra channels
//   Y : [B][Cout][H][W]
// 32x64 tile per wave; branch-free K loop (9 taps x Cp/4 steps, 8 WMMA each).
// ---------------------------------------------------------------------------
__global__ __launch_bounds__(32)
void conv3x3_kernel(const float* __restrict__ Wt_,
                    const float* __restrict__ Xp_,
                    float* __restrict__ Y_,
                    int Cout, int Cp, int Bn, int H, int Wd)
{
    const int lane = threadIdx.x & 31;
    const int half = lane >> 4;
    const int l16  = lane & 15;
    const int HW = H * Wd, N = Bn * HW;
    const int Wp = Wd + 2;
    const long HpWp = (long)(H + 2) * Wp;
    const int n0 = blockIdx.x * 64;
    const int m0 = blockIdx.y * 32;

    gcf Wt = (gcf)Wt_;
    gcf Xp = (gcf)Xp_;
    gf  Y  = (gf)Y_;

    gcf xb[4];
#pragma unroll
    for (int t = 0; t < 4; ++t) {
        const int col = min(n0 + t * 16 + l16, N - 1);     // clamped
        const int b = col / HW;
        const int rem = col - b * HW;
        const int y = rem / Wd;
        const int x = rem - y * Wd;
        xb[t] = Xp + (size_t)b * Cp * HpWp + (size_t)y * Wp + x
                   + (size_t)(half * 2) * HpWp;
    }
    gcf w0 = Wt + (size_t)min(m0 + l16,      Cout - 1) * 9 * Cp + half * 2;
    gcf w1 = Wt + (size_t)min(m0 + 16 + l16, Cout - 1) * 9 * Cp + half * 2;

    v8f acc[8] = {};
    for (int kk = 0; kk < 9; ++kk) {
        const int ky = kk / 3;
        const int kx = kk - ky * 3;
        const long soff = (long)ky * Wp + kx;
        gcf xo0 = xb[0] + soff;
        gcf xo1 = xb[1] + soff;
        gcf xo2 = xb[2] + soff;
        gcf xo3 = xb[3] + soff;
        gcf wp0 = w0 + kk * Cp;
        gcf wp1 = w1 + kk * Cp;
        for (int c0 = 0; c0 < Cp; c0 += 4) {
            v2f a0 = *(gcv2f)wp0;      // contiguous channel pair -> global_load_b64
            v2f a1 = *(gcv2f)wp1;
            v2f b0, b1, b2, b3;
            b0.x = xo0[0]; b0.y = xo0[HpWp];
            b1.x = xo1[0]; b1.y = xo1[HpWp];
            b2.x = xo2[0]; b2.y = xo2[HpWp];
            b3.x = xo3[0]; b3.y = xo3[HpWp];
            acc[0] = wmma4(a0, b0, acc[0]);
            acc[1] = wmma4(a0, b1, acc[1]);
            acc[2] = wmma4(a0, b2, acc[2]);
            acc[3] = wmma4(a0, b3, acc[3]);
            acc[4] = wmma4(a1, b0, acc[4]);
            acc[5] = wmma4(a1, b1, acc[5]);
            acc[6] = wmma4(a1, b2, acc[6]);
            acc[7] = wmma4(a1, b3, acc[7]);
            wp0 += 4; wp1 += 4;
            xo0 += 4 * HpWp; xo1 += 4 * HpWp; xo2 += 4 * HpWp; xo3 += 4 * HpWp;
        }
    }

#pragma unroll
    for (int t = 0; t < 4; ++t) {
        const int col = n0 + t * 16 + l16;
        if (col >= N) continue;
        const int b = col / HW;
        const int rem = col - b * HW;
#pragma unroll
        for (int g = 0; g < 2; ++g)
#pragma unroll
            for (int r = 0; r < 8; ++r) {
                const int row = m0 + g * 16 + r + half * 8;
                if (row < Cout)
                    Y[((size_t)(b * Cout + row)) * HW + rem] = acc[g * 4 + t][r];
            }
    }
}

// zero-bordered, channel-padded input copy
__global__ void pad_input_kernel(const float* __restrict__ X, float* __restrict__ Xp,
                                 int Bn, int Cin, int Cp, int H, int W)
{
    const int Hp = H + 2, Wp = W + 2;
    long total = (long)Bn * Cp * Hp * Wp;
    long i = (long)blockIdx.x * blockDim.x + threadIdx.x;
    if (i >= total) return;
    const int x = (int)(i % Wp);
    long t = i / Wp;
    const int y = (int)(t % Hp); t /= Hp;
    const int c = (int)(t % Cp);
    const int b = (int)(t / Cp);
    float v = 0.f;
    if (c < Cin && y >= 1 && y <= H && x >= 1 && x <= W)
        v = X[((long)(b * Cin + c) * H + (y - 1)) * W + (x - 1)];
    Xp[i] = v;
}

// weights [Cout][Cin][3][3] -> [Cout][9][Cp] (tap-major, channel-inner, padded)
__global__ void pad_weight_kernel(const float* __restrict__ W, float* __restrict__ Wp,
                                  int Cout, int Cin, int Cp)
{
    long total = (long)Cout * 9 * Cp;
    long i = (long)blockIdx.x * blockDim.x + threadIdx.x;
    if (i >= total) return;
    const int ci = (int)(i % Cp);
    long t = i / Cp;
    const int kk = (int)(t % 9);
    const int m = (int)(t / 9);
    Wp[i] = (ci < Cin) ? W[((long)m * Cin + ci) * 9 + kk] : 0.f;
}

// ---------------------------------------------------------------------------
// BatchNorm (training stats, biased var) + ReLU ; elementwise & misc kernels
// ---------------------------------------------------------------------------
__global__ void bn_stats_kernel(const float* __restrict__ X, float* __restrict__ mean,
                                float* __restrict__ var, int C, int Bn, int HW)
{
    const int c = blockIdx.x;
    const int tid = threadIdx.x;
    const int n = Bn * HW;
    double s = 0.0, s2 = 0.0;
    for (int i = tid; i < n; i += blockDim.x) {
        const int b = i / HW;
        const int r = i - b * HW;
        const float v = X[((long)(b * C + c)) * HW + r];
        s += v; s2 += (double)v * v;
    }
    __shared__ double sh[256], sh2[256];
    sh[tid] = s; sh2[tid] = s2;
    __syncthreads();
    for (int off = blockDim.x >> 1; off > 0; off >>= 1) {
        if (tid < off) { sh[tid] += sh[tid + off]; sh2[tid] += sh2[tid + off]; }
        __syncthreads();
    }
    if (tid == 0) {
        const double m = sh[0] / n;
        mean[c] = (float)m;
        var[c]  = (float)(sh2[0] / n - m * m);
    }
}

__global__ void bn_relu_kernel(float* __restrict__ X, const float* __restrict__ mean,
                               const float* __restrict__ var, const float* __restrict__ g,
                               const float* __restrict__ b, int C, int HW, long total)
{
    long i = (long)blockIdx.x * blockDim.x + threadIdx.x;
    if (i >= total) return;
    const int c = (int)((i / HW) % C);
    const float v = (X[i] - mean[c]) * rsqrtf(var[c] + 1e-5f) * g[c] + b[c];
    X[i] = fmaxf(v, 0.f);
}

__global__ void maxpool2_kernel(const float* __restrict__ X, float* __restrict__ Y,
                                int BC, int H, int W)
{
    const int Ho = H / 2, Wo = W / 2;
    long total = (long)BC * Ho * Wo;
    long i = (long)blockIdx.x * blockDim.x + threadIdx.x;
    if (i >= total) return;
    const int bc = (int)(i / (Ho * Wo));
    const int r = (int)(i - (long)bc * Ho * Wo);
    const int y = r / Wo, x = r - y * Wo;
    const float* p = X + ((long)bc * H + 2 * y) * W + 2 * x;
    Y[i] = fmaxf(fmaxf(p[0], p[1]), fmaxf(p[W], p[W + 1]));
}

// bilinear 2x upsample (align_corners) + zero pad, into channel slice of concat
__global__ void upsample_cat_kernel(const float* __restrict__ X, float* __restrict__ D,
                                    int Bn, int C, int Hs, int Ws, int Ctot, int coff,
                                    int Ho, int Wo)
{
    long total = (long)Bn * C * Ho * Wo;
    long i = (long)blockIdx.x * blockDim.x + threadIdx.x;
    if (i >= total) return;
    const int x = (int)(i % Wo);
    long t = i / Wo;
    const int y = (int)(t % Ho); t /= Ho;
    const int c = (int)(t % C);
    const int b = (int)(t / C);
    const int Hu = 2 * Hs, Wu = 2 * Ws;
    float v = 0.f;
    if (y < Hu && x < Wu) {   // pad bottom/right with zeros (dY//2 == 0)
        const float py = y * (float)(Hs - 1) / (float)(Hu - 1);
        const float px = x * (float)(Ws - 1) / (float)(Wu - 1);
        int y0 = (int)floorf(py), x0 = (int)floorf(px);
        const int y1 = min(y0 + 1, Hs - 1), x1 = min(x0 + 1, Ws - 1);
        const float wy = py - y0, wx = px - x0;
        const float* S = X + ((long)(b * C + c)) * Hs * Ws;
        const float v00 = S[y0 * Ws + x0], v01 = S[y0 * Ws + x1];
        const float v10 = S[y1 * Ws + x0], v11 = S[y1 * Ws + x1];
        v = (v00 * (1.f - wx) + v01 * wx) * (1.f - wy)
          + (v10 * (1.f - wx) + v11 * wx) * wy;
    }
    D[((long)(b * Ctot + coff + c) * Ho + y) * Wo + x] = v;
}

__global__ void copy_cat_kernel(const float* __restrict__ X, float* __restrict__ D,
                                int Bn, int C, int HW, int Ctot, int coff)
{
    long total = (long)Bn * C * HW;
    long i = (long)blockIdx.x * blockDim.x + threadIdx.x;
    if (i >= total) return;
    const int r = (int)(i % HW);
    long t = i / HW;
    const int c = (int)(t % C);
    const int b = (int)(t / C);
    D[((long)(b * Ctot + coff + c)) * HW + r] = X[i];
}

__global__ void norm_kernel(const float* __restrict__ e, float* __restrict__ nrm,
                            int rows, int E)
{
    int r = blockIdx.x * blockDim.x + threadIdx.x;
    if (r >= rows) return;
    float s = 0.f;
    const float* p = e + (long)r * E;
    for (int i = 0; i < E; ++i) s += p[i] * p[i];
    nrm[r] = sqrtf(s);
}

// fm[b][c][y][x] = { sim1[b,x,y], sim2[b,x,y], sim3[b, n=x, o=y] }
__global__ void build_fm_kernel(const float* __restrict__ sim1, const float* __restrict__ nrm,
                                const float* __restrict__ sim3, float* __restrict__ fm,
                                int Bn, int N)
{
    long total = (long)Bn * 3 * N * N;
    long i = (long)blockIdx.x * blockDim.x + threadIdx.x;
    if (i >= total) return;
    const int x = (int)(i % N);
    long t = i / N;
    const int y = (int)(t % N); t /= N;
    const int c = (int)(t % 3);
    const int b = (int)(t / 3);
    float v;
    if (c == 0) {
        v = sim1[((long)b * N + x) * N + y];
    } else if (c == 1) {
        const float s = sim1[((long)b * N + x) * N + y];
        const float d = fmaxf(nrm[b * N + x] * nrm[b * N + y], 1e-6f);
        v = s / d;
    } else {
        v = sim3[((long)(b * N + x)) * N + y];   // sim3 stored [row=(b,n)][o]
    }
    fm[i] = v;
}

// z_s[p,c]=tanh(ps[b,i,c]+attn[p,c]); z_o[p,c]=tanh(po[b,k,c]+attn[p,c]); p=(b,k,i)
__global__ void zbuild_kernel(const float* __restrict__ attn, const float* __restrict__ ps,
                              const float* __restrict__ po, float* __restrict__ zs,
                              float* __restrict__ zo, int Bn, int N, int C)
{
    long total = (long)Bn * N * N * C;
    long idx = (long)blockIdx.x * blockDim.x + threadIdx.x;
    if (idx >= total) return;
    const int c = (int)(idx % C);
    long p = idx / C;
    const int i = (int)(p % N);
    long bk = p / N;
    const int k = (int)(bk % N);
    const int b = (int)(bk / N);
    const float a = attn[idx];
    zs[idx] = tanhf(ps[((long)(b * N + i)) * C + c] + a);
    zo[idx] = tanhf(po[((long)(b * N + k)) * C + c] + a);
}

// ---------------------------------------------------------------------------
// Orchestration
// ---------------------------------------------------------------------------
static inline dim3 eg(long total, int bs = 256) { return dim3((unsigned)CDIV(total, (long)bs)); }

extern "C" void kernel_launch(void* const* d_in, const int* in_sizes, int n_in,
                              void* d_out, int out_size, void* d_ws, size_t ws_size,
                              hipStream_t stream)
{
    (void)in_sizes; (void)n_in; (void)out_size; (void)ws_size;
    const int Bn = 4, N = 42, E = 768;
    const int ROWS = Bn * N;            // 168
    const int PAIRS = Bn * N * N;       // 7056
    const int HW42 = 42 * 42, HW21 = 21 * 21, HW10 = 10 * 10;

    const float* e       = (const float*)d_in[0];
    const float* bil3_w  = (const float*)d_in[1];
    const float* inc_w1  = (const float*)d_in[2];
    const float* inc_g1  = (const float*)d_in[3];
    const float* inc_b1  = (const float*)d_in[4];
    const float* inc_w2  = (const float*)d_in[5];
    const float* inc_g2  = (const float*)d_in[6];
    const float* inc_b2  = (const float*)d_in[7];
    const float* d1_w1   = (const float*)d_in[8];
    const float* d1_g1   = (const float*)d_in[9];
    const float* d1_b1   = (const float*)d_in[10];
    const float* d1_w2   = (const float*)d_in[11];
    const float* d1_g2   = (const float*)d_in[12];
    const float* d1_b2   = (const float*)d_in[13];
    const float* d2_w1   = (const float*)d_in[14];
    const float* d2_g1   = (const float*)d_in[15];
    const float* d2_b1   = (const float*)d_in[16];
    const float* d2_w2   = (const float*)d_in[17];
    const float* d2_g2   = (const float*)d_in[18];
    const float* d2_b2   = (const float*)d_in[19];
    const float* u1_w1   = (const float*)d_in[20];
    const float* u1_g1   = (const float*)d_in[21];
    const float* u1_b1   = (const float*)d_in[22];
    const float* u1_w2   = (const float*)d_in[23];
    const float* u1_g2   = (const float*)d_in[24];
    const float* u1_b2   = (const float*)d_in[25];
    const float* u2_w1   = (const float*)d_in[26];
    const float* u2_g1   = (const float*)d_in[27];
    const float* u2_b1   = (const float*)d_in[28];
    const float* u2_w2   = (const float*)d_in[29];
    const float* u2_g2   = (const float*)d_in[30];
    const float* u2_b2   = (const float*)d_in[31];
    const float* outc_w  = (const float*)d_in[32];
    const float* outc_b  = (const float*)d_in[33];
    const float* lin_w   = (const float*)d_in[34];
    const float* lin_b   = (const float*)d_in[35];
    const float* ws_w    = (const float*)d_in[36];
    const float* wo_w    = (const float*)d_in[37];
    const float* bil2_w  = (const float*)d_in[38];
    const float* bil2_b  = (const float*)d_in[39];
    float* out = (float*)d_out;

    // ---- bump arena in d_ws --------------------------------------------
    char* wsp = (char*)d_ws;
    auto alloc = [&](size_t nfloat) -> float* {
        float* p = (float*)wsp;
        wsp += ((nfloat * sizeof(float) + 255) / 256) * 256;
        return p;
    };
    float* nrm  = alloc(ROWS);
    float* sim1 = alloc((size_t)Bn * N * N);
    float* sim3 = alloc((size_t)ROWS * N);
    float* P3   = alloc((size_t)N * ROWS * 12);
    float* fm   = alloc((size_t)Bn * 3 * HW42);
    float* bnm  = alloc(1024);
    float* bnv  = alloc(1024);
    float* xpb  = alloc((size_t)Bn * 512 * 44 * 44);     // padded conv input (max use)
    float* wpb  = alloc((size_t)256 * 9 * 1024);         // padded/reordered weights (max)
    float* ta   = alloc((size_t)Bn * 256 * HW42);        // conv scratch
    float* x1   = alloc((size_t)Bn * 256 * HW42);
    float* pb   = alloc((size_t)Bn * 256 * HW21);
    float* x2   = alloc((size_t)Bn * 512 * HW21);
    float* x3   = alloc((size_t)Bn * 512 * HW10);
    float* cat1 = alloc((size_t)Bn * 1024 * HW21);
    float* xu1  = alloc((size_t)Bn * 256 * HW21);
    float* cat2 = alloc((size_t)Bn * 512 * HW42);
    float* xu2  = alloc((size_t)Bn * 128 * HW42);
    float* oc   = alloc((size_t)Bn * 256 * HW42);
    float* attn = alloc((size_t)PAIRS * 256);
    float* psb  = alloc((size_t)ROWS * 256);
    float* pob  = alloc((size_t)ROWS * 256);
    float* zs   = alloc((size_t)PAIRS * 256);
    float* zo   = alloc((size_t)PAIRS * 256);
    float* P2   = alloc((size_t)97 * PAIRS * 4);

    // ---- similarity features -------------------------------------------
    norm_kernel<<<eg(ROWS, 64), 64, 0, stream>>>(e, nrm, ROWS, E);

    // sim1[b] = e[b] (42x768) @ e[b]^T
    gemm_f32_kernel<<<dim3(CDIV(N, 64), CDIV(N, 16), Bn), 32, 0, stream>>>(
        e, E, 1, (long)N * E,
        e, 1, E, (long)N * E,
        sim1, N, (long)N * N,
        N, N, E, nullptr, 0);

    // sim3: for each o, T = Emat(168x768) @ W3[o](768x768); rowdot with Emat
    gemm_rowdot_kernel<<<dim3(12, CDIV(ROWS, 32), N), 32, 0, stream>>>(
        e, E, bil3_w, (long)E * E, e, E, P3, ROWS, E, E, 12);
    reduce_p_kernel<<<eg((long)N * ROWS), 256, 0, stream>>>(P3, sim3, N, ROWS, 12, nullptr);

    build_fm_kernel<<<eg((long)Bn * 3 * HW42), 256, 0, stream>>>(sim1, nrm, sim3, fm, Bn, N);

    // ---- UNet -----------------------------------------------------------
    auto conv_bn_relu = [&](const float* w, const float* g, const float* bb,
                            const float* X, float* Y, int Cout, int Cin, int H, int W) {
        const int Cp = (Cin + 3) & ~3;
        const int Hp = H + 2, Wpd = W + 2;
        pad_input_kernel<<<eg((long)Bn * Cp * Hp * Wpd), 256, 0, stream>>>(
            X, xpb, Bn, Cin, Cp, H, W);
        pad_weight_kernel<<<eg((long)Cout * 9 * Cp), 256, 0, stream>>>(
            w, wpb, Cout, Cin, Cp);
        const int n = Bn * H * W;
        conv3x3_kernel<<<dim3(CDIV(n, 64), CDIV(Cout, 32)), 32, 0, stream>>>(
            wpb, xpb, Y, Cout, Cp, Bn, H, W);
        bn_stats_kernel<<<Cout, 256, 0, stream>>>(Y, bnm, bnv, Cout, Bn, H * W);
        const long total = (long)Bn * Cout * H * W;
        bn_relu_kernel<<<eg(total), 256, 0, stream>>>(Y, bnm, bnv, g, bb, Cout, H * W, total);
    };

    // inc: fm(3ch,42) -> x1(256ch,42)
    conv_bn_relu(inc_w1, inc_g1, inc_b1, fm, ta, 256, 3, 42, 42);
    conv_bn_relu(inc_w2, inc_g2, inc_b2, ta, x1, 256, 256, 42, 42);
    // down1
    maxpool2_kernel<<<eg((long)Bn * 256 * HW21), 256, 0, stream>>>(x1, pb, Bn * 256, 42, 42);
    conv_bn_relu(d1_w1, d1_g1, d1_b1, pb, ta, 512, 256, 21, 21);
    conv_bn_relu(d1_w2, d1_g2, d1_b2, ta, x2, 512, 512, 21, 21);
    // down2
    maxpool2_kernel<<<eg((long)Bn * 512 * HW10), 256, 0, stream>>>(x2, pb, Bn * 512, 21, 21);
    conv_bn_relu(d2_w1, d2_g1, d2_b1, pb, ta, 512, 512, 10, 10);
    conv_bn_relu(d2_w2, d2_g2, d2_b2, ta, x3, 512, 512, 10, 10);
    // up1: cat[x2, up(x3)+pad] (1024ch,21) -> 256 -> 256
    copy_cat_kernel<<<eg((long)Bn * 512 * HW21), 256, 0, stream>>>(x2, cat1, Bn, 512, HW21, 1024, 0);
    upsample_cat_kernel<<<eg((long)Bn * 512 * HW21), 256, 0, stream>>>(
        x3, cat1, Bn, 512, 10, 10, 1024, 512, 21, 21);
    conv_bn_relu(u1_w1, u1_g1, u1_b1, cat1, ta, 256, 1024, 21, 21);
    conv_bn_relu(u1_w2, u1_g2, u1_b2, ta, xu1, 256, 256, 21, 21);
    // up2: cat[x1, up(xu1)] (512ch,42) -> 128 -> 128
    copy_cat_kernel<<<eg((long)Bn * 256 * HW42), 256, 0, stream>>>(x1, cat2, Bn, 256, HW42, 512, 0);
    upsample_cat_kernel<<<eg((long)Bn * 256 * HW42), 256, 0, stream>>>(
        xu1, cat2, Bn, 256, 21, 21, 512, 256, 42, 42);
    conv_bn_relu(u2_w1, u2_g1, u2_b1, cat2, ta, 128, 512, 42, 42);
    conv_bn_relu(u2_w2, u2_g2, u2_b2, ta, xu2, 128, 128, 42, 42);

    // outc (1x1 conv, per-M bias): oc[b] = outc_w(256x128) @ xu2[b](128x1764) + b
    gemm_f32_kernel<<<dim3(CDIV(HW42, 64), CDIV(256, 16), Bn), 32, 0, stream>>>(
        outc_w, 128, 1, 0,
        xu2, HW42, 1, (long)128 * HW42,
        oc, HW42, (long)256 * HW42,
        256, HW42, 128, outc_b, 1);

    // linear on NHWC attn: attn[b] = oc[b]^T(1764x256) @ lin_w^T + lin_b
    gemm_f32_kernel<<<dim3(CDIV(256, 64), CDIV(HW42, 16), Bn), 32, 0, stream>>>(
        oc, 1, HW42, (long)256 * HW42,
        lin_w, 1, 256, 0,
        attn, 256, (long)HW42 * 256,
        HW42, 256, 256, lin_b, 2);

    // ps = e @ ws^T ; po = e @ wo^T   (168 x 256)
    gemm_f32_kernel<<<dim3(CDIV(256, 64), CDIV(ROWS, 16), 1), 32, 0, stream>>>(
        e, E, 1, 0, ws_w, 1, E, 0, psb, 256, 0, ROWS, 256, E, nullptr, 0);
    gemm_f32_kernel<<<dim3(CDIV(256, 64), CDIV(ROWS, 16), 1), 32, 0, stream>>>(
        e, E, 1, 0, wo_w, 1, E, 0, pob, 256, 0, ROWS, 256, E, nullptr, 0);

    zbuild_kernel<<<eg((long)PAIRS * 256), 256, 0, stream>>>(attn, psb, pob, zs, zo, Bn, N, 256);

    // final bilinear head: for each o (97): T = zs(7056x256) @ W2[o](256x256); rowdot zo
    gemm_rowdot_kernel<<<dim3(4, CDIV(PAIRS, 32), 97), 32, 0, stream>>>(
        zs, 256, bil2_w, (long)256 * 256, zo, 256, P2, PAIRS, 256, 256, 4);
    reduce_p_kernel<<<eg((long)97 * PAIRS), 256, 0, stream>>>(P2, out, 97, PAIRS, 4, bil2_b);
}